// Regrid_24936580120740
// MI455X (gfx1250) — compile-verified
//
#include <hip/hip_runtime.h>
#include <cstdint>
#include <cstddef>

// ---- problem constants (ERA5 0.25deg -> 1deg regrid) ----
#define NB   65160      // destination cells (181*360)
#define NA   1038240    // source cells (721*1440)
#define NXS  1440       // source longitude
#define NYS  721        // source latitude (flipped axis)

typedef __attribute__((ext_vector_type(2))) float v2f;
typedef __attribute__((ext_vector_type(8))) float v8f;

// ------------------------------------------------------------------
// Phase 1a: zero the histogram + scatter cursors
// ------------------------------------------------------------------
__global__ void k_zero(int* __restrict__ p, int n) {
  int i = blockIdx.x * blockDim.x + threadIdx.x;
  if (i < n) p[i] = 0;
}

// ------------------------------------------------------------------
// Phase 1b: histogram of destination rows
// ------------------------------------------------------------------
__global__ void k_hist(const int* __restrict__ rows, int* __restrict__ counts, int nnz) {
  int k = blockIdx.x * blockDim.x + threadIdx.x;
  if (k < nnz) atomicAdd(&counts[rows[k]], 1);
}

// ------------------------------------------------------------------
// Phase 1c: exclusive scan -> row_ptr[NB+1] (single 1024-thread block,
// Hillis-Steele per 1024-chunk with running carry)
// ------------------------------------------------------------------
__global__ void k_scan(const int* __restrict__ counts, int* __restrict__ row_ptr) {
  __shared__ int sm[1024];
  __shared__ int s_run;
  const int tid = threadIdx.x;
  if (tid == 0) s_run = 0;
  __syncthreads();
  for (int base = 0; base < NB; base += 1024) {
    int i = base + tid;
    int v = (i < NB) ? counts[i] : 0;
    sm[tid] = v;
    __syncthreads();
    for (int off = 1; off < 1024; off <<= 1) {
      int t = (tid >= off) ? sm[tid - off] : 0;
      __syncthreads();
      sm[tid] += t;
      __syncthreads();
    }
    int incl = sm[tid];
    int run  = s_run;
    __syncthreads();                       // everyone has read s_run
    if (i < NB) row_ptr[i] = run + incl - v;
    if (tid == 1023) s_run = run + incl;   // carry = chunk total
    __syncthreads();
  }
  if (tid == 0) row_ptr[NB] = s_run;       // == nnz
}

// ------------------------------------------------------------------
// Phase 1d: scatter entries into CSR order, 16B-padded entries.
// Latitude flip folded into the stored column index (one-time integer
// work instead of per-gather work in the hot kernel).
// Entry (int4) = {dest_row, flipped_col, bitcast(weight), 0}
// ------------------------------------------------------------------
__global__ void k_scatter(const int* __restrict__ rows, const int* __restrict__ cols,
                          const float* __restrict__ wts, const int* __restrict__ row_ptr,
                          int* __restrict__ cursor, int4* __restrict__ ent, int nnz) {
  int k = blockIdx.x * blockDim.x + threadIdx.x;
  if (k >= nnz) return;
  int b   = rows[k];
  int pos = row_ptr[b] + atomicAdd(&cursor[b], 1);
  int col = cols[k];
  int r   = col / NXS;                       // compiler lowers to mul-shift
  int c   = col - r * NXS;
  int fc  = (NYS - 1 - r) * NXS + c;         // jnp.flip(axis=-2)
  ent[pos] = make_int4(b, fc, __float_as_int(wts[k]), 0);
}

// ------------------------------------------------------------------
// Phase 2: tiled SpMM on the matrix pipe, exact f32.
//   D(16 n x 16 b) += A(16x4) @ B(4x16)  via V_WMMA_F32_16X16X4_F32
//   A[n_local][K] = w_K * x_flipped[n0+n_local, col_K]   (gather, f32)
//   B[K][b_local] = (row(entry K) == b0+b_local) ? 1 : 0 (exact 0/1)
// Out-of-range K slots are handled entirely on the B side: their
// indicator column is all-zero, so A needs NO mask (clamped index reads
// a valid finite entry; 0 * finite = 0). This keeps the entry fetch an
// unconditional b96 and the loop free of EXEC manipulation.
// One wave32 per 16x16 output tile; 2x unroll with independent
// accumulator chains so both chunks' loads overlap.
// A-layout (32-bit 16x4): lanes 0-15 K=0,1 ; lanes 16-31 K=2,3 (per VGPR)
// B-layout (32-bit 4x16): lanes 0-15 K=0,1 ; lanes 16-31 K=2,3
// D-layout (f32 16x16)  : VGPR v -> M = v + 8*(lane>=16), N = lane&15
// ------------------------------------------------------------------
__global__ __launch_bounds__(32)
void k_spmm_wmma(const float* __restrict__ x, const int* __restrict__ row_ptr,
                 const int4* __restrict__ ent, float* __restrict__ out, int nch) {
  const int tb = blockIdx.x, tn = blockIdx.y;
  const int b0 = tb * 16, n0 = tn * 16;
  const int l    = threadIdx.x;
  const int half = l >> 4;        // 0: K=0,1   1: K=2,3
  const int lo   = l & 15;        // A: n_local ; B: b_local

  const int b_hi = (b0 + 16 < NB) ? b0 + 16 : NB;
  const int s = row_ptr[b0];
  const int e = row_ptr[b_hi];

  const int ng    = n0 + lo;
  const int n_eff = (ng < nch) ? ng : (nch - 1);        // clamp: tail rows unstored
  const float* __restrict__ xrow = x + (size_t)n_eff * NA;

  v8f acc0 = {}, acc1 = {};
  for (int k0 = s; k0 < e; k0 += 8) {                   // uniform across the wave
#pragma unroll
    for (int u = 0; u < 2; ++u) {
      int g0 = k0 + 4 * u + half * 2;
      int g1 = g0 + 1;
      int i0 = (g0 < e) ? g0 : (e - 1);                 // clamp to a valid entry
      int i1 = (g1 < e) ? g1 : (e - 1);
      int4 E0 = ent[i0];                                // unconditional wide load
      int4 E1 = ent[i1];

      v2f A, B;
      A.x = __int_as_float(E0.z) * xrow[E0.y];          // no mask needed on A
      A.y = __int_as_float(E1.z) * xrow[E1.y];
      B.x = (g0 < e && E0.x == b0 + lo) ? 1.0f : 0.0f;  // exact row indicator
      B.y = (g1 < e && E1.x == b0 + lo) ? 1.0f : 0.0f;

      if (u == 0)
        acc0 = __builtin_amdgcn_wmma_f32_16x16x4_f32(
            false, A, false, B, (short)0, acc0, false, false);
      else
        acc1 = __builtin_amdgcn_wmma_f32_16x16x4_f32(
            false, A, false, B, (short)0, acc1, false, false);
    }
  }
  v8f acc = acc0 + acc1;

  // Epilogue: coalesced (16 consecutive b per half-wave), atomic-free.
  const int bg = b0 + lo;
  const bool full = (b0 + 16 <= NB) && (n0 + 16 <= nch);  // wave-uniform
  if (full) {
    float* __restrict__ o = out + (size_t)(n0 + half * 8) * NB + bg;
#pragma unroll
    for (int v = 0; v < 8; ++v) o[(size_t)v * NB] = acc[v];
  } else {
#pragma unroll
    for (int v = 0; v < 8; ++v) {
      int ngl = n0 + v + half * 8;
      if (ngl < nch && bg < NB)
        out[(size_t)ngl * NB + bg] = acc[v];
    }
  }
}

// ------------------------------------------------------------------
extern "C" void kernel_launch(void* const* d_in, const int* in_sizes, int n_in,
                              void* d_out, int out_size, void* d_ws, size_t ws_size,
                              hipStream_t stream) {
  const float* x    = (const float*)d_in[0];
  const int*   rows = (const int*)d_in[1];
  const int*   cols = (const int*)d_in[2];
  const float* wts  = (const float*)d_in[3];
  const int nnz = in_sizes[1];
  const int nch = in_sizes[0] / NA;      // 2*70 = 140 leading channels

  // workspace carve (~5 MB total); ent padded to 16B alignment
  int* counts  = (int*)d_ws;                       // [NB]
  int* cursor  = counts + NB;                      // [NB]
  int* row_ptr = cursor + NB;                      // [NB+1]
  int  ent_off = (2 * NB + NB + 1 + 3) & ~3;       // round to int4 boundary
  int4* ent    = (int4*)(counts + ent_off);        // [nnz] {row, col, w_bits, 0}

  k_zero<<<(2 * NB + 255) / 256, 256, 0, stream>>>(counts, 2 * NB);
  k_hist<<<(nnz + 255) / 256, 256, 0, stream>>>(rows, counts, nnz);
  k_scan<<<1, 1024, 0, stream>>>(counts, row_ptr);
  k_scatter<<<(nnz + 255) / 256, 256, 0, stream>>>(rows, cols, wts, row_ptr,
                                                   cursor, ent, nnz);

  dim3 grid((NB + 15) / 16, (nch + 15) / 16);   // 4073 x 9 tiles, 1 wave each
  k_spmm_wmma<<<grid, 32, 0, stream>>>(x, row_ptr, ent, (float*)d_out, nch);
}